// ImprovedDecoder_67912022884409
// MI455X (gfx1250) — compile-verified
//
#include <hip/hip_runtime.h>
#include <hip/hip_bf16.h>

#define HID 128
#define LATENT 256
#define SP 130   // padded LDS row stride (floats) -> conflict-free ds_load_b64

typedef __attribute__((ext_vector_type(2))) float v2f;
typedef __attribute__((ext_vector_type(8))) float v8f;

// ---------------------------------------------------------------------------
// Kernel 1: latent decoder  g = relu(LN(z@ld_w1^T + b1)) @ ld_w2^T + b2
// ---------------------------------------------------------------------------
__global__ void latent_kernel(const float* __restrict__ z,
                              const float* __restrict__ w1,
                              const float* __restrict__ b1,
                              const float* __restrict__ gam,
                              const float* __restrict__ bet,
                              const float* __restrict__ w2,
                              const float* __restrict__ b2,
                              float* __restrict__ gout) {
  __shared__ float sz[LATENT];
  __shared__ float sh[2 * HID];
  __shared__ float red[2 * HID];
  int t = threadIdx.x;            // 0..255
  sz[t] = z[t];
  __syncthreads();
  float acc = b1[t];
  #pragma unroll 8
  for (int k = 0; k < LATENT; ++k) acc += sz[k] * w1[t * LATENT + k];
  red[t] = acc; __syncthreads();
  for (int s = 128; s > 0; s >>= 1) { if (t < s) red[t] += red[t + s]; __syncthreads(); }
  float mu = red[0] * (1.0f / 256.0f);
  __syncthreads();
  float d = acc - mu;
  red[t] = d * d; __syncthreads();
  for (int s = 128; s > 0; s >>= 1) { if (t < s) red[t] += red[t + s]; __syncthreads(); }
  float var = red[0] * (1.0f / 256.0f);
  float y = d * rsqrtf(var + 1e-5f) * gam[t] + bet[t];
  sh[t] = fmaxf(y, 0.0f);
  __syncthreads();
  if (t < HID) {
    float a2 = b2[t];
    #pragma unroll 8
    for (int k = 0; k < 2 * HID; ++k) a2 += sh[k] * w2[t * (2 * HID) + k];
    gout[t] = a2;
  }
}

// ---------------------------------------------------------------------------
// Kernel 2: build combined = [node_embeddings | broadcast(g)]  (N x 256)
// ---------------------------------------------------------------------------
__global__ void build_combined(const float* __restrict__ emb,
                               const float* __restrict__ g,
                               float* __restrict__ comb) {
  int i = blockIdx.x, t = threadIdx.x;  // 256 threads
  comb[i * 256 + t] = (t < HID) ? emb[i * HID + t] : g[t - HID];
}

// ---------------------------------------------------------------------------
// Kernel 3: generic fp32 WMMA GEMM:  C[M,N] = A[M,K](lda) @ W[N,K](ldb)^T + bias
// One wave per 16x16 C tile, V_WMMA_F32_16X16X4_F32 along K.
// ---------------------------------------------------------------------------
__global__ void gemm_wmma_f32(const float* __restrict__ A, int lda,
                              const float* __restrict__ W, int ldb,
                              const float* __restrict__ bias,
                              float* __restrict__ C,
                              int M, int N, int K) {
  int lane = threadIdx.x & 31;
  int wid  = (blockIdx.x * blockDim.x + threadIdx.x) >> 5;
  int tn = N >> 4, tm = M >> 4;
  if (wid >= tm * tn) return;                 // whole-wave exit: EXEC stays uniform
  int tile_m = wid / tn, tile_n = wid % tn;

  int mrow = tile_m * 16 + (lane & 15);
  int ncol = tile_n * 16 + (lane & 15);
  int koff = (lane >> 4) << 1;                // 0 for lanes 0-15, 2 for 16-31

  const float* aptr = A + mrow * lda + koff;
  const float* bptr = W + ncol * ldb + koff;

  v8f acc = {};
  #pragma unroll 8
  for (int k0 = 0; k0 < K; k0 += 4) {
    v2f a = *(const v2f*)(aptr + k0);         // 8-byte aligned
    v2f b = *(const v2f*)(bptr + k0);
    acc = __builtin_amdgcn_wmma_f32_16x16x4_f32(
        /*neg_a=*/false, a, /*neg_b=*/false, b,
        /*c_mod=*/(short)0, acc, /*reuse_a=*/false, /*reuse_b=*/false);
  }

  float bv = bias ? bias[ncol] : 0.0f;
  int rbase = tile_m * 16 + ((lane >> 4) << 3);   // +8 for upper half-wave
  #pragma unroll
  for (int v = 0; v < 8; ++v) {
    C[(rbase + v) * N + ncol] = acc[v] + bv;
  }
}

// ---------------------------------------------------------------------------
// Kernel 4: per-row LayerNorm(128) + relu
// ---------------------------------------------------------------------------
__global__ void ln_relu_128(const float* __restrict__ x,
                            const float* __restrict__ gam,
                            const float* __restrict__ bet,
                            float* __restrict__ out) {
  __shared__ float red[HID];
  int row = blockIdx.x, t = threadIdx.x;  // 128 threads
  float v = x[row * HID + t];
  red[t] = v; __syncthreads();
  for (int s = 64; s > 0; s >>= 1) { if (t < s) red[t] += red[t + s]; __syncthreads(); }
  float mu = red[0] * (1.0f / 128.0f);
  __syncthreads();
  float d = v - mu;
  red[t] = d * d; __syncthreads();
  for (int s = 64; s > 0; s >>= 1) { if (t < s) red[t] += red[t + s]; __syncthreads(); }
  float var = red[0] * (1.0f / 128.0f);
  out[row * HID + t] = fmaxf(d * rsqrtf(var + 1e-5f) * gam[t] + bet[t], 0.0f);
}

// ---------------------------------------------------------------------------
// Kernel 5: symmetric edge scores, upper-triangular blocks only.
// Block (bi,bj) with bi<=bj computes s(i,j) and s(j,i), writes BOTH
// out[i,j] and out[j,i] = 0.5*(sigmoid(s_ij)+sigmoid(s_ji)).
// pi already includes em_b1.  LDS rows padded to SP=130 floats so the 16
// lanes with distinct tj hit distinct banks (2*tj) on ds_load_b64.
// ---------------------------------------------------------------------------
__global__ void edge_kernel(const float* __restrict__ piB,
                            const float* __restrict__ pjB,
                            const float* __restrict__ w,
                            const float* __restrict__ b2,
                            float* __restrict__ out, int N) {
  __shared__ float sPiI[16 * SP];  // pi rows of i-range
  __shared__ float sPjJ[16 * SP];  // pj rows of j-range
  __shared__ float sPiJ[16 * SP];  // pi rows of j-range (for s_ji)
  __shared__ float sPjI[16 * SP];  // pj rows of i-range (for s_ji)
  __shared__ float sW[HID];

  int bi = blockIdx.y, bj = blockIdx.x;
  if (bi > bj) return;             // lower triangle handled by transpose write

  int t = threadIdx.x;  // 256 threads
  for (int idx = t; idx < 16 * HID; idx += 256) {
    int r = idx >> 7, h = idx & (HID - 1);
    sPiI[r * SP + h] = piB[(bi * 16 + r) * HID + h];
    sPjJ[r * SP + h] = pjB[(bj * 16 + r) * HID + h];
    sPiJ[r * SP + h] = piB[(bj * 16 + r) * HID + h];
    sPjI[r * SP + h] = pjB[(bi * 16 + r) * HID + h];
  }
  if (t < HID) sW[t] = w[t];
  __syncthreads();

  int ti = t >> 4, tj = t & 15;
  const v2f* pI = (const v2f*)&sPiI[ti * SP];
  const v2f* pJ = (const v2f*)&sPjJ[tj * SP];
  const v2f* qJ = (const v2f*)&sPiJ[tj * SP];
  const v2f* qI = (const v2f*)&sPjI[ti * SP];
  const v2f* wV = (const v2f*)sW;

  v2f a1 = {0.0f, 0.0f}, a2 = {0.0f, 0.0f};
  #pragma unroll 8
  for (int h2 = 0; h2 < HID / 2; ++h2) {
    v2f wv = wV[h2];
    v2f x = pI[h2] + pJ[h2];       // hidden(i,j,·) pre-relu
    v2f y = qJ[h2] + qI[h2];       // hidden(j,i,·) pre-relu
    x.x = fmaxf(x.x, 0.0f); x.y = fmaxf(x.y, 0.0f);
    y.x = fmaxf(y.x, 0.0f); y.y = fmaxf(y.y, 0.0f);
    a1 += x * wv;
    a2 += y * wv;
  }
  float s1 = a1.x + a1.y;
  float s2 = a2.x + a2.y;

  float bb = b2[0];
  float e1 = 1.0f / (1.0f + __expf(-(s1 + bb)));
  float e2 = 1.0f / (1.0f + __expf(-(s2 + bb)));
  float val = 0.5f * (e1 + e2);    // symmetric in (i,j)

  int i = bi * 16 + ti, j = bj * 16 + tj;
  out[i * N + j] = val;
  out[j * N + i] = val;            // diagonal tile: duplicate identical write
}

// ---------------------------------------------------------------------------
extern "C" void kernel_launch(void* const* d_in, const int* in_sizes, int n_in,
                              void* d_out, int out_size, void* d_ws, size_t ws_size,
                              hipStream_t stream) {
  const float* z     = (const float*)d_in[0];
  const float* emb   = (const float*)d_in[1];
  // d_in[2] = num_nodes (int); derive from emb size instead
  const float* ld_w1 = (const float*)d_in[3];
  const float* ld_b1 = (const float*)d_in[4];
  const float* ld_g  = (const float*)d_in[5];
  const float* ld_be = (const float*)d_in[6];
  const float* ld_w2 = (const float*)d_in[7];
  const float* ld_b2 = (const float*)d_in[8];
  const float* nr_w1 = (const float*)d_in[9];
  const float* nr_b1 = (const float*)d_in[10];
  const float* nr_g  = (const float*)d_in[11];
  const float* nr_be = (const float*)d_in[12];
  const float* nr_w2 = (const float*)d_in[13];
  const float* nr_b2 = (const float*)d_in[14];
  const float* em_w1 = (const float*)d_in[15];
  const float* em_b1 = (const float*)d_in[16];
  const float* em_w2 = (const float*)d_in[17];
  const float* em_b2 = (const float*)d_in[18];

  int N = in_sizes[1] / HID;   // 1024

  float* ws   = (float*)d_ws;
  float* g    = ws;                      // 128 (padded to 256)
  float* comb = ws + 256;                // N*256
  float* h1   = comb + (size_t)N * 256;  // N*128
  float* h1r  = h1   + (size_t)N * HID;
  float* nf   = h1r  + (size_t)N * HID;
  float* pi   = nf   + (size_t)N * HID;
  float* pj   = pi   + (size_t)N * HID;

  // 1) latent decoder -> g
  latent_kernel<<<1, 256, 0, stream>>>(z, ld_w1, ld_b1, ld_g, ld_be, ld_w2, ld_b2, g);

  // 2) combined = [emb | g]
  build_combined<<<N, 256, 0, stream>>>(emb, g, comb);

  int tiles  = (N / 16) * (HID / 16);
  int blocks = (tiles + 3) / 4;          // 4 waves (128 thr) per block

  // 3) h1 = combined @ nr_w1^T + nr_b1          (M=N, K=256, N=128)
  gemm_wmma_f32<<<blocks, 128, 0, stream>>>(comb, 256, nr_w1, 256, nr_b1, h1, N, HID, 256);

  // 4) h1r = relu(LN(h1))
  ln_relu_128<<<N, 128, 0, stream>>>(h1, nr_g, nr_be, h1r);

  // 5) node_feat = h1r @ nr_w2^T + nr_b2        (K=128)
  gemm_wmma_f32<<<blocks, 128, 0, stream>>>(h1r, 128, nr_w2, 128, nr_b2, nf, N, HID, 128);

  // 6) pi = node_feat @ Wa^T + em_b1   (Wa = em_w1[:, :128], row stride 256)
  gemm_wmma_f32<<<blocks, 128, 0, stream>>>(nf, 128, em_w1, 256, em_b1, pi, N, HID, 128);

  // 7) pj = node_feat @ Wb^T           (Wb = em_w1[:, 128:], row stride 256)
  gemm_wmma_f32<<<blocks, 128, 0, stream>>>(nf, 128, em_w1 + HID, 256, nullptr, pj, N, HID, 128);

  // 8) symmetric sigmoid edge matrix (upper-triangular blocks, mirrored writes)
  dim3 egrid(N / 16, N / 16);
  edge_kernel<<<egrid, 256, 0, stream>>>(pi, pj, em_w2, em_b2, (float*)d_out, N);
}